// LinearSISR_78116865179900
// MI455X (gfx1250) — compile-verified
//
#include <hip/hip_runtime.h>

// Fully-fused LinearSISR head for MI455X (gfx1250, wave32, WMMA).
//
// Roofline: materializing reg/reg_up/mix would move ~1.5 GB (~65us @ 23.3TB/s).
// Fusing everything per 16x16 output tile cuts global traffic to ~45 MB (~2us),
// so the ~14 GFLOP of compute (dominated by the [pixels,72]x[72,25] mixing
// GEMM) becomes the limiter -> map it onto V_WMMA_F32_16X16X4_F32 (K=72 = 18*4
// exactly; fp32 WMMA preserves reference precision).
//
// Per 256-thread block (8 waves) / 16x16 hi-res pixel tile:
//   1) stage x halo (3x14x14), conv weights, basis K in LDS
//   2) compute 3x20x20 bicubic patch in LDS (reflect pad + clamped taps fused)
//   3) pipelined k-loop: double-buffered 10x10x4ch "reg" conv blocks in LDS;
//      each wave fuses the x2 bilinear (0.25/0.75 parity weights) into the
//      WMMA A-fragment and accumulates 2 rowgroups x 2 tapgroups
//   4) scatter mix[256x25] to LDS per D-matrix layout
//   5) 25-tap spatially-varying conv, 3 channels, write out (mix never in HBM)

#define BB   8
#define CC   3
#define HH   256
#define WW   256
#define DD   72
#define HO   512
#define KS   5
#define NT   25
#define TIL  16

typedef float v2f __attribute__((ext_vector_type(2)));
typedef float v8f __attribute__((ext_vector_type(8)));

__device__ __forceinline__ float cubicw(float d) {
    const float Aa = -0.75f;
    float ad = fabsf(d);
    float ad2 = ad * ad, ad3 = ad2 * ad;
    float w1 = (Aa + 2.f) * ad3 - (Aa + 3.f) * ad2 + 1.f;
    float w2 = Aa * ad3 - 5.f * Aa * ad2 + 8.f * Aa * ad - 4.f * Aa;
    return ad <= 1.f ? w1 : (ad < 2.f ? w2 : 0.f);
}

__device__ __forceinline__ int clampi(int v, int lo, int hi) {
    return v < lo ? lo : (v > hi ? hi : v);
}

__global__ __launch_bounds__(256)
void sisr_fused(const float* __restrict__ x, const float* __restrict__ cw,
                const float* __restrict__ cb, const float* __restrict__ Kb,
                float* __restrict__ out)
{
    __shared__ float s_x[CC][14][14];     // low-res x halo (zero-padded OOB for conv)
    __shared__ float s_w[DD * 27];        // conv weights
    __shared__ float s_b[DD];             // conv bias
    __shared__ float s_K[DD * NT];        // mixing basis [72][25]
    __shared__ float s_reg[2][10 * 10 * 4]; // double-buffered 4-channel reg block
    __shared__ float s_mix[256 * NT];     // per-pixel mixing kernels [256][25]
    __shared__ float s_bic[CC][20][20];   // bicubic x2 patch incl. 5x5 halo

    const int tid = threadIdx.x;
    const int b   = blockIdx.z;
    const int H0  = blockIdx.y * TIL;     // hi-res tile origin
    const int W0  = blockIdx.x * TIL;
    const int XB  = (H0 >> 1) - 3;        // x-tile row base (covers conv + bicubic taps)
    const int XWB = (W0 >> 1) - 3;
    const int LB  = (H0 >> 1) - 1;        // reg-tile row base (bilinear halo)
    const int LWB = (W0 >> 1) - 1;

    // ---------- phase 1: cooperative LDS loads ----------
    for (int i = tid; i < CC * 14 * 14; i += 256) {
        int c = i / 196, r = (i % 196) / 14, col = i % 14;
        int gr = XB + r, gc = XWB + col;
        float v = 0.f;
        if (gr >= 0 && gr < HH && gc >= 0 && gc < WW)
            v = x[((size_t)(b * CC + c) * HH + gr) * WW + gc];
        s_x[c][r][col] = v;
    }
    for (int i = tid; i < DD * 27; i += 256) s_w[i] = cw[i];
    for (int i = tid; i < DD;      i += 256) s_b[i] = cb[i];
    for (int i = tid; i < DD * NT; i += 256) s_K[i] = Kb[i];
    __syncthreads();

    // 3x3 conv for one 4-channel block of reg at clamped low-res coords
    auto reg_block = [&](int kb, int buf) {
        for (int i = tid; i < 400; i += 256) {       // i = (rr*10+cc)*4+dd
            int dd = i & 3;
            int d  = kb * 4 + dd;
            int cc2 = (i >> 2) % 10, rr = i / 40;
            int gr = clampi(LB + rr,  0, HH - 1);
            int gc = clampi(LWB + cc2, 0, WW - 1);
            float acc = s_b[d];
            #pragma unroll
            for (int c = 0; c < CC; ++c)
                #pragma unroll
                for (int ki = 0; ki < 3; ++ki)
                    #pragma unroll
                    for (int kj = 0; kj < 3; ++kj)
                        acc += s_x[c][gr - 1 + ki - XB][gc - 1 + kj - XWB] *
                               s_w[((d * CC + c) * 3 + ki) * 3 + kj];
            s_reg[buf][i] = acc;
        }
    };

    // ---------- phase 2: bicubic x2 patch (reflect pad fused) + reg block 0 ----------
    for (int i = tid; i < CC * 20 * 20; i += 256) {
        int c = i / 400, rem = i % 400, hp = rem / 20, wp = rem % 20;
        int h2 = H0 + hp - 2; h2 = h2 < 0 ? -h2 : (h2 > HO - 1 ? 2 * (HO - 1) - h2 : h2);
        int w2 = W0 + wp - 2; w2 = w2 < 0 ? -w2 : (w2 > HO - 1 ? 2 * (HO - 1) - w2 : w2);
        float sy = 0.5f * (float)h2 - 0.25f;
        float sxx = 0.5f * (float)w2 - 0.25f;
        int by = (int)floorf(sy), bx = (int)floorf(sxx);
        float wy[4], wx[4]; int ly[4], lx[4];
        #pragma unroll
        for (int t = 0; t < 4; ++t) {
            int ty = by - 1 + t, tx = bx - 1 + t;
            wy[t] = cubicw(sy - (float)ty);
            wx[t] = cubicw(sxx - (float)tx);
            ly[t] = clampi(ty, 0, HH - 1) - XB;   // clamped taps stay inside valid tile
            lx[t] = clampi(tx, 0, WW - 1) - XWB;
        }
        float v = 0.f;
        #pragma unroll
        for (int ti2 = 0; ti2 < 4; ++ti2) {
            float rv = 0.f;
            #pragma unroll
            for (int tj2 = 0; tj2 < 4; ++tj2) rv += wx[tj2] * s_x[c][ly[ti2]][lx[tj2]];
            v += wy[ti2] * rv;
        }
        s_bic[c][hp][wp] = v;
    }
    reg_block(0, 0);
    __syncthreads();

    // ---------- phase 3: pipelined WMMA mixing-GEMM ----------
    const int  lane = tid & 31;
    const int  wv   = tid >> 5;         // wave id, each wave owns 2 pixel rows
    const int  m    = lane & 15;        // A row (px) / B,D column (tap)
    const bool hi   = lane >= 16;
    const int  dsel = hi ? 2 : 0;       // K-slot pair per ISA 16x4 f32 A layout

    // bilinear fetch coeffs for this lane's A rows (fixed 0.25/0.75 by parity)
    int   i00[2]; float wq[2][4];
    #pragma unroll
    for (int rg2 = 0; rg2 < 2; ++rg2) {
        int py = wv * 2 + rg2;
        int ly0 = (py + (py & 1)) >> 1;
        float wyl = (py & 1) ? 0.75f : 0.25f;
        int px = m;
        int lx0 = (px + (px & 1)) >> 1;
        float wxl = (px & 1) ? 0.75f : 0.25f;
        float wyh = 1.f - wyl, wxh = 1.f - wxl;
        i00[rg2] = (ly0 * 10 + lx0) * 4;
        wq[rg2][0] = wyl * wxl; wq[rg2][1] = wyl * wxh;
        wq[rg2][2] = wyh * wxl; wq[rg2][3] = wyh * wxh;
    }

    v8f acc[2][2] = {};                 // [rowgroup][tapgroup]
    for (int kb = 0; kb < 18; ++kb) {   // K = 72 = 18 * 4
        if (kb < 17) reg_block(kb + 1, (kb + 1) & 1);   // produce next block
        const float* rb = s_reg[kb & 1];                 // consume current block
        #pragma unroll
        for (int rg2 = 0; rg2 < 2; ++rg2) {
            int base = i00[rg2];
            v2f a;   // fused bilinear upsample -> A fragment (2 K-slots per lane)
            a.x = wq[rg2][0] * rb[base + dsel]      + wq[rg2][1] * rb[base + 4 + dsel]
                + wq[rg2][2] * rb[base + 40 + dsel] + wq[rg2][3] * rb[base + 44 + dsel];
            a.y = wq[rg2][0] * rb[base + dsel + 1]      + wq[rg2][1] * rb[base + 4 + dsel + 1]
                + wq[rg2][2] * rb[base + 40 + dsel + 1] + wq[rg2][3] * rb[base + 44 + dsel + 1];
            #pragma unroll
            for (int tg = 0; tg < 2; ++tg) {
                int q = tg * 16 + m;
                int d0 = kb * 4 + dsel;
                v2f bf;  // B fragment: basis K[d, q], zero-padded taps 25..31
                bf.x = (q < NT) ? s_K[d0 * NT + q]       : 0.f;
                bf.y = (q < NT) ? s_K[(d0 + 1) * NT + q] : 0.f;
                acc[rg2][tg] = __builtin_amdgcn_wmma_f32_16x16x4_f32(
                    false, a, false, bf, (short)0, acc[rg2][tg], false, false);
            }
        }
        __syncthreads();
    }

    // ---------- phase 4: scatter mix to LDS (D layout: VGPR r -> row r(+8)) ----------
    #pragma unroll
    for (int rg2 = 0; rg2 < 2; ++rg2) {
        int py = wv * 2 + rg2;
        #pragma unroll
        for (int tg = 0; tg < 2; ++tg) {
            int tap = tg * 16 + m;
            if (tap < NT) {
                #pragma unroll
                for (int r = 0; r < 8; ++r) {
                    int px = r + (hi ? 8 : 0);
                    s_mix[(py * 16 + px) * NT + tap] = acc[rg2][tg][r];
                }
            }
        }
    }
    __syncthreads();

    // ---------- phase 5: spatially-varying 5x5 conv, write output ----------
    {
        int p = tid, py = p >> 4, px = p & 15;
        float a0 = 0.f, a1 = 0.f, a2 = 0.f;
        #pragma unroll
        for (int ti2 = 0; ti2 < KS; ++ti2)
            #pragma unroll
            for (int tj2 = 0; tj2 < KS; ++tj2) {
                float mv = s_mix[p * NT + ti2 * KS + tj2];
                a0 += mv * s_bic[0][py + ti2][px + tj2];
                a1 += mv * s_bic[1][py + ti2][px + tj2];
                a2 += mv * s_bic[2][py + ti2][px + tj2];
            }
        int h2 = H0 + py, w2 = W0 + px;
        size_t o0 = ((size_t)(b * CC) * HO + h2) * HO + w2;
        out[o0]                        = a0;
        out[o0 + (size_t)HO * HO]      = a1;
        out[o0 + 2 * (size_t)HO * HO]  = a2;
    }
}

extern "C" void kernel_launch(void* const* d_in, const int* in_sizes, int n_in,
                              void* d_out, int out_size, void* d_ws, size_t ws_size,
                              hipStream_t stream) {
    (void)in_sizes; (void)n_in; (void)out_size; (void)d_ws; (void)ws_size;
    const float* x  = (const float*)d_in[0];
    const float* cw = (const float*)d_in[1];
    const float* cb = (const float*)d_in[2];
    const float* Kb = (const float*)d_in[3];
    dim3 grid(HO / TIL, HO / TIL, BB);   // 32 x 32 x 8 tiles
    sisr_fused<<<grid, dim3(256), 0, stream>>>(x, cw, cb, Kb, (float*)d_out);
}